// AttributeEncoder_85993835200641
// MI455X (gfx1250) — compile-verified
//
#include <hip/hip_runtime.h>
#include <stdint.h>

// ---------------------------------------------------------------------------
// AttributeEncoder scatter-add on MI455X (gfx1250, wave32)
//
// Counting-sort events by destination cell, then a WMMA-based segmented sum
// (V_WMMA_F32_16X16X4_F32) that writes every output cell exactly once.
// No float atomics; the 128MB output and 4MB table live in the 192MB L2.
// ---------------------------------------------------------------------------

#define CELLS   131072      // B*W*H*L = 4*32*32*32
#define EMB_D   256
#define GROUP   16          // cells per workgroup in the WMMA pass
#define NGROUPS (CELLS / GROUP)
#define TILE    512         // metadata slots staged in LDS per iteration

typedef __attribute__((ext_vector_type(2))) float v2f;
typedef __attribute__((ext_vector_type(8))) float v8f;

__global__ void ae_zero_u32(unsigned int* __restrict__ p, int n) {
  int i = blockIdx.x * blockDim.x + threadIdx.x;
  if (i < n) p[i] = 0u;
}

// Pass 1: histogram of events per cell (u32 atomics only).
__global__ void ae_count(const int* __restrict__ cell,
                         unsigned int* __restrict__ counts, int n) {
  int i = blockIdx.x * blockDim.x + threadIdx.x;
  if (i < n) atomicAdd(&counts[cell[i]], 1u);
}

// Pass 2: exclusive scan over 131072 counts -> offsets[CELLS+1] and a
// mutable cursor copy. Single workgroup, 1024 threads x 128 elements each.
__global__ __launch_bounds__(1024)
void ae_scan(const unsigned int* __restrict__ counts,
             unsigned int* __restrict__ offsets,
             unsigned int* __restrict__ cursor) {
  __shared__ unsigned int sums[1024];
  const int t    = threadIdx.x;
  const int PER  = CELLS / 1024;       // 128
  const int base = t * PER;
  unsigned int s = 0;
  for (int i = 0; i < PER; ++i) s += counts[base + i];
  sums[t] = s;
  __syncthreads();
  // Kogge-Stone inclusive scan over the 1024 partial sums.
  for (int off = 1; off < 1024; off <<= 1) {
    unsigned int v = (t >= off) ? sums[t - off] : 0u;
    __syncthreads();
    sums[t] += v;
    __syncthreads();
  }
  unsigned int run = sums[t] - s;      // exclusive prefix for this thread
  for (int i = 0; i < PER; ++i) {
    offsets[base + i] = run;
    cursor [base + i] = run;
    run += counts[base + i];
  }
  if (t == 1023) offsets[CELLS] = sums[1023];
}

// Pass 3: scatter events into cell-sorted order. row < 4096 (12b),
// cell < 131072 (17b) -> pack into one u32: (row << 17) | cell.
__global__ void ae_scatter(const int* __restrict__ cell,
                           const int* __restrict__ row,
                           unsigned int* __restrict__ cursor,
                           unsigned int* __restrict__ packed, int n) {
  int i = blockIdx.x * blockDim.x + threadIdx.x;
  if (i < n) {
    int c = cell[i];
    unsigned int pos = atomicAdd(&cursor[c], 1u);
    packed[pos] = ((unsigned int)row[i] << 17) | (unsigned int)c;
  }
}

// Pass 4: segmented sum via WMMA. One workgroup = 16 cells (M), 16 waves
// each owning a 16-column tile (N) of D=256. K iterates over the group's
// concatenated sorted events in chunks of 4; A is the block-diagonal
// one-hot event->cell selector, B the gathered table rows.
//
// Segment metadata is staged in LDS (512 slots/tile) so the 16 waves share
// one coalesced global load, and the hot loop is branch-free: invalid slots
// carry sentinel 0x80000000, whose row extract maps to a safe row 0 while
// the one-hot A entry is forced to 0 via v_cmp/v_cndmask (EXEC stays all-1s
// for WMMA).
//
// f32 WMMA operand layout (ISA 7.12.2):
//   A 16x4:  lane L holds row M=L&15; VGPR0 = K=2*(L>>4), VGPR1 = K+1
//   B 4x16:  lane L holds col N=L&15; VGPR0 = K=2*(L>>4), VGPR1 = K+1
//   C/D:     lane L holds col N=L&15; VGPR v = row M = v + 8*(L>>4)
__global__ __launch_bounds__(512)
void ae_segsum_wmma(const float* __restrict__ table,
                    const unsigned int* __restrict__ offsets,
                    const unsigned int* __restrict__ packed,
                    float* __restrict__ out) {
  __shared__ unsigned int smeta[TILE];

  const int cellBase = blockIdx.x * GROUP;
  const unsigned int segBase = offsets[cellBase];
  const unsigned int total   = offsets[cellBase + GROUP] - segBase;

  const int lane    = threadIdx.x & 31;
  const int wave    = threadIdx.x >> 5;     // 0..15 -> column tile
  const int colBase = wave * 16;
  const int mn      = lane & 15;            // A-row M / B-col N for this lane
  const int h       = lane >> 4;            // K-half selector
  const unsigned int target = (unsigned int)(cellBase + mn);  // cell this lane selects

  v8f acc = {};
  for (unsigned int tileBase = 0; tileBase < total; tileBase += TILE) {
    // Stage up to TILE packed (row|cell) words; pad with sentinel.
    {
      const unsigned int s = tileBase + threadIdx.x;
      smeta[threadIdx.x] = (s < total) ? packed[segBase + s] : 0x80000000u;
    }
    __syncthreads();

    const unsigned int cnt    = min((unsigned int)TILE, total - tileBase);
    const unsigned int nchunk = (cnt + 3u) >> 2;
    for (unsigned int c = 0; c < nchunk; ++c) {
      const unsigned int k0 = 4u * c + 2u * (unsigned int)h;  // K = 2h
      const unsigned int p0 = smeta[k0];                      // ds_load (broadcast)
      const unsigned int p1 = smeta[k0 + 1];

      const bool ok0 = p0 < 0x80000000u;
      const bool ok1 = p1 < 0x80000000u;
      const unsigned int r0 = (p0 >> 17) & 0xFFFu;  // sentinel -> row 0 (safe)
      const unsigned int r1 = (p1 >> 17) & 0xFFFu;

      v2f a, b;
      a.x = (ok0 && (p0 & 0x1FFFFu) == target) ? 1.0f : 0.0f;  // one-hot A
      a.y = (ok1 && (p1 & 0x1FFFFu) == target) ? 1.0f : 0.0f;
      b.x = table[(size_t)r0 * EMB_D + colBase + mn];          // gathered B
      b.y = table[(size_t)r1 * EMB_D + colBase + mn];

      // D = A x B + C, full fp32 accumulate
      acc = __builtin_amdgcn_wmma_f32_16x16x4_f32(
          /*neg_a=*/false, a, /*neg_b=*/false, b,
          /*c_mod=*/(short)0, acc, /*reuse_a=*/false, /*reuse_b=*/false);
    }
    __syncthreads();
  }

  // Write D: each (cell, col) is written exactly once across the grid, so
  // the poisoned output buffer is fully overwritten (zeros for empty cells).
  const int col = colBase + mn;
#pragma unroll
  for (int v = 0; v < 8; ++v) {
    const int m = v + 8 * h;
    out[(size_t)(cellBase + m) * EMB_D + col] = acc[v];
  }
}

extern "C" void kernel_launch(void* const* d_in, const int* in_sizes, int n_in,
                              void* d_out, int out_size, void* d_ws, size_t ws_size,
                              hipStream_t stream) {
  const float* table      = (const float*)d_in[0];
  const int*   event_cell = (const int*)d_in[1];
  const int*   event_row  = (const int*)d_in[2];
  const int    n          = in_sizes[1];
  float*       out        = (float*)d_out;

  // Workspace layout (u32): counts[CELLS] | offsets[CELLS+1] | cursor[CELLS]
  //                         | packed[n]   (~5.8 MB total)
  unsigned int* counts  = (unsigned int*)d_ws;
  unsigned int* offsets = counts + CELLS;
  unsigned int* cursor  = offsets + (CELLS + 1);
  unsigned int* packed  = cursor + CELLS;

  ae_zero_u32<<<(CELLS + 255) / 256, 256, 0, stream>>>(counts, CELLS);
  ae_count  <<<(n + 255) / 256, 256, 0, stream>>>(event_cell, counts, n);
  ae_scan   <<<1, 1024, 0, stream>>>(counts, offsets, cursor);
  ae_scatter<<<(n + 255) / 256, 256, 0, stream>>>(event_cell, event_row, cursor,
                                                  packed, n);
  ae_segsum_wmma<<<NGROUPS, 512, 0, stream>>>(table, offsets, packed, out);
}